// GraphAppnp_33767032881494
// MI455X (gfx1250) — compile-verified
//
#include <hip/hip_runtime.h>
#include <hip/hip_bf16.h>

typedef __attribute__((ext_vector_type(2))) float v2f;
typedef __attribute__((ext_vector_type(8))) float v8f;

#define KCHUNK   16
#define LDSTRIDE 132   // 128 + 4 pad to dodge bank conflicts on transposed fills

// ---------------------------------------------------------------------------
// Wave-tile geometry: 256 threads = 8 waves (wave32), wave grid 2(M) x 4(N).
// Wave tile 64(M) x 32(N) = 4 x 2 subtiles of 16x16, each a v8f accumulator.
// LDS chunks are K-major: As[k][m], Bs[k][n] so fragment loads are lane-stride-1.
// Double-buffered (ping-pong) LDS with register staging: next chunk's global
// loads are issued BEFORE the WMMA block so the 32 back-to-back v_wmma ops
// hide the HBM/L2 latency; registers are committed to the other LDS buffer
// after the math, with a single barrier per K iteration.
//
// V_WMMA_F32_16X16X4_F32 fragment layout (ISA 7.12.2):
//   A 16x4 : lane%16 = M ; component c holds K = 2*(lane/16) + c
//   B 4x16 : lane%16 = N ; component c holds K(row) = 2*(lane/16) + c
//   C 16x16: VGPR v -> row v + 8*(lane/16), col = lane%16
// ---------------------------------------------------------------------------

__device__ __forceinline__ void wmma_block(const float (*As)[LDSTRIDE],
                                           const float (*Bs)[LDSTRIDE],
                                           int wm, int wn, int half, int l16,
                                           v8f acc[4][2]) {
#pragma unroll
  for (int kk = 0; kk < KCHUNK; kk += 4) {
    const int ka = kk + 2 * half;
    v2f afrag[4], bfrag[2];
#pragma unroll
    for (int ms = 0; ms < 4; ++ms) {
      const int m = wm * 64 + ms * 16 + l16;
      afrag[ms].x = As[ka][m];
      afrag[ms].y = As[ka + 1][m];
    }
#pragma unroll
    for (int ns = 0; ns < 2; ++ns) {
      const int nn = wn * 32 + ns * 16 + l16;
      bfrag[ns].x = Bs[ka][nn];
      bfrag[ns].y = Bs[ka + 1][nn];
    }
#pragma unroll
    for (int ms = 0; ms < 4; ++ms)
#pragma unroll
      for (int ns = 0; ns < 2; ++ns)
        acc[ms][ns] = __builtin_amdgcn_wmma_f32_16x16x4_f32(
            false, afrag[ms], false, bfrag[ns], (short)0, acc[ms][ns],
            false, false);
  }
}

__device__ __forceinline__ void store_chunk(float (*S)[LDSTRIDE], int kq,
                                            int jm, const float r[8]) {
#pragma unroll
  for (int i = 0; i < 8; ++i) S[kq + i][jm] = r[i];
}

// ---------------------------------------------------------------------------
// Kernel 1: circular-pad 3x3 conv as implicit GEMM.
//   Out[b, co, p] = sum_k W[co, k] * Xpatch[b, k, p] + bias[co]
//   M = 256 (co), N = 3136 (pixels), K = 576 (ci*9), 36 K-chunks.
// ---------------------------------------------------------------------------
__device__ __forceinline__ void conv_loadA(const float* __restrict__ Wt,
                                           int m0, int jm, int kq, int k0,
                                           float r[8]) {
  const float* src = Wt + (size_t)(m0 + jm) * 576 + k0 + kq;
#pragma unroll
  for (int i = 0; i < 8; ++i) r[i] = src[i];
}

__device__ __forceinline__ void conv_loadB(const float* __restrict__ Xb,
                                           int ph, int pw, bool pvalid,
                                           int kq, int k0, float r[8]) {
#pragma unroll
  for (int i = 0; i < 8; ++i) {
    const int k  = k0 + kq + i;
    const int ci = k / 9;
    const int kr = k - ci * 9;
    const int rr = kr / 3;
    const int cc = kr - rr * 3;
    int sh = ph + rr - 1;
    sh += (sh < 0) ? 56 : 0;
    sh -= (sh >= 56) ? 56 : 0;
    int sw = pw + cc - 1;
    sw += (sw < 0) ? 56 : 0;
    sw -= (sw >= 56) ? 56 : 0;
    r[i] = pvalid ? Xb[ci * 3136 + sh * 56 + sw] : 0.0f;
  }
}

__global__ __launch_bounds__(256) void conv_wmma(
    const float* __restrict__ X, const float* __restrict__ Wt,
    const float* __restrict__ bias, float* __restrict__ Out) {
  __shared__ float As[2][KCHUNK][LDSTRIDE];
  __shared__ float Bs[2][KCHUNK][LDSTRIDE];

  const int b  = blockIdx.z;
  const int m0 = blockIdx.y * 128;   // out-channel base
  const int n0 = blockIdx.x * 128;   // pixel base
  const int tid  = threadIdx.x;
  const int lane = tid & 31;
  const int wave = tid >> 5;
  const int half = lane >> 4;
  const int l16  = lane & 15;
  const int wm = wave & 1;
  const int wn = wave >> 1;

  v8f acc[4][2];
#pragma unroll
  for (int i = 0; i < 4; ++i)
#pragma unroll
    for (int j = 0; j < 2; ++j) acc[i][j] = (v8f)0.0f;

  const float* Xb = X + (size_t)b * 64 * 3136;

  const int jm = tid & 127;          // column (m or pixel-in-tile)
  const int kq = (tid >> 7) * 8;     // 8 consecutive K rows per thread

  const int p  = n0 + jm;
  const int ph = p / 56;
  const int pw = p - ph * 56;
  const bool pvalid = (p < 3136);

  float ra[8], rb[8];
  conv_loadA(Wt, m0, jm, kq, 0, ra);
  conv_loadB(Xb, ph, pw, pvalid, kq, 0, rb);
  store_chunk(As[0], kq, jm, ra);
  store_chunk(Bs[0], kq, jm, rb);
  __syncthreads();

  int cur = 0;
  const int NC = 576 / KCHUNK;  // 36
  for (int c = 0; c < NC; ++c) {
    const bool more = (c + 1 < NC);
    if (more) {
      conv_loadA(Wt, m0, jm, kq, (c + 1) * KCHUNK, ra);
      conv_loadB(Xb, ph, pw, pvalid, kq, (c + 1) * KCHUNK, rb);
    }
    wmma_block(As[cur], Bs[cur], wm, wn, half, l16, acc);
    if (more) {
      store_chunk(As[cur ^ 1], kq, jm, ra);
      store_chunk(Bs[cur ^ 1], kq, jm, rb);
    }
    __syncthreads();
    cur ^= 1;
  }

  float* Ob = Out + (size_t)b * 256 * 3136;
#pragma unroll
  for (int ms = 0; ms < 4; ++ms)
#pragma unroll
    for (int ns = 0; ns < 2; ++ns) {
      const int col = n0 + wn * 32 + ns * 16 + l16;
      if (col < 3136) {
#pragma unroll
        for (int v = 0; v < 8; ++v) {
          const int row = m0 + wm * 64 + ms * 16 + v + 8 * half;
          Ob[(size_t)row * 3136 + col] = acc[ms][ns][v] + bias[row];
        }
      }
    }
}

// ---------------------------------------------------------------------------
// Kernel 2: attention scores a[b] = h1f[b] (256x3136) x hnf[b]^T (3136x256).
// Both operands row-major [256][3136]; both tiles transpose-on-load into
// K-major LDS. K = 3136 = 196 * 16 exactly.
// ---------------------------------------------------------------------------
__global__ __launch_bounds__(256) void scores_wmma(
    const float* __restrict__ H1, const float* __restrict__ HN,
    float* __restrict__ A) {
  __shared__ float As[2][KCHUNK][LDSTRIDE];
  __shared__ float Bs[2][KCHUNK][LDSTRIDE];

  const int b  = blockIdx.z;
  const int m0 = blockIdx.y * 128;
  const int n0 = blockIdx.x * 128;
  const int tid  = threadIdx.x;
  const int lane = tid & 31;
  const int wave = tid >> 5;
  const int half = lane >> 4;
  const int l16  = lane & 15;
  const int wm = wave & 1;
  const int wn = wave >> 1;

  v8f acc[4][2];
#pragma unroll
  for (int i = 0; i < 4; ++i)
#pragma unroll
    for (int j = 0; j < 2; ++j) acc[i][j] = (v8f)0.0f;

  const float* h1b = H1 + (size_t)b * 256 * 3136;
  const float* hnb = HN + (size_t)b * 256 * 3136;

  const int jm = tid & 127;
  const int kq = (tid >> 7) * 8;

  const float* pa = h1b + (size_t)(m0 + jm) * 3136 + kq;
  const float* pb = hnb + (size_t)(n0 + jm) * 3136 + kq;

  float ra[8], rb[8];
#pragma unroll
  for (int i = 0; i < 8; ++i) ra[i] = pa[i];
#pragma unroll
  for (int i = 0; i < 8; ++i) rb[i] = pb[i];
  store_chunk(As[0], kq, jm, ra);
  store_chunk(Bs[0], kq, jm, rb);
  __syncthreads();

  int cur = 0;
  const int NC = 3136 / KCHUNK;  // 196
  for (int c = 0; c < NC; ++c) {
    const bool more = (c + 1 < NC);
    if (more) {
      const int kn = (c + 1) * KCHUNK;
#pragma unroll
      for (int i = 0; i < 8; ++i) ra[i] = pa[kn + i];
#pragma unroll
      for (int i = 0; i < 8; ++i) rb[i] = pb[kn + i];
    }
    wmma_block(As[cur], Bs[cur], wm, wn, half, l16, acc);
    if (more) {
      store_chunk(As[cur ^ 1], kq, jm, ra);
      store_chunk(Bs[cur ^ 1], kq, jm, rb);
    }
    __syncthreads();
    cur ^= 1;
  }

  float* Ab = A + (size_t)b * 256 * 256;
#pragma unroll
  for (int ms = 0; ms < 4; ++ms)
#pragma unroll
    for (int ns = 0; ns < 2; ++ns) {
      const int col = n0 + wn * 32 + ns * 16 + l16;
#pragma unroll
      for (int v = 0; v < 8; ++v) {
        const int row = m0 + wm * 64 + ms * 16 + v + 8 * half;
        Ab[row * 256 + col] = acc[ms][ns][v];
      }
    }
}

// ---------------------------------------------------------------------------
// Kernel 3: LeakyReLU + softmax over axis=1 (columns of the 256x256 matrix).
// One block per batch; thread m owns column m; three coalesced passes.
// ---------------------------------------------------------------------------
__global__ __launch_bounds__(256) void softmax_col(float* __restrict__ A) {
  const int b = blockIdx.x;
  const int m = threadIdx.x;
  float* Ab = A + (size_t)b * 256 * 256;

  float mx = -3.402823466e38f;
  for (int n = 0; n < 256; ++n) {
    float v = Ab[n * 256 + m];
    v = (v >= 0.0f) ? v : 0.01f * v;
    mx = fmaxf(mx, v);
  }
  float s = 0.0f;
  for (int n = 0; n < 256; ++n) {
    float v = Ab[n * 256 + m];
    v = (v >= 0.0f) ? v : 0.01f * v;
    s += __expf(v - mx);
  }
  const float inv = 1.0f / s;
  for (int n = 0; n < 256; ++n) {
    float v = Ab[n * 256 + m];
    v = (v >= 0.0f) ? v : 0.01f * v;
    Ab[n * 256 + m] = __expf(v - mx) * inv;
  }
}

// ---------------------------------------------------------------------------
// Kernel 4: agg = attn[b] (256x256) x h1f[b] (256x3136); fused epilogue
//   out = 0.1*agg + 0.9*h1.  M=256 (n), N=3136 (f), K=256 (m), 16 K-chunks.
// B tile is row-major direct (no transpose); A tile transpose-on-load.
// ---------------------------------------------------------------------------
__device__ __forceinline__ void agg_loadB(const float* __restrict__ Hb,
                                          int n0, int br, int bc, int k0,
                                          float r[8]) {
  const float* s2 = Hb + (size_t)(k0 + br) * 3136 + n0 + bc;
#pragma unroll
  for (int i = 0; i < 8; ++i)
    r[i] = (n0 + bc + i < 3136) ? s2[i] : 0.0f;
}

__global__ __launch_bounds__(256) void agg_wmma(
    const float* __restrict__ Attn, const float* __restrict__ H1,
    float* __restrict__ Out) {
  __shared__ float As[2][KCHUNK][LDSTRIDE];
  __shared__ float Bs[2][KCHUNK][LDSTRIDE];

  const int b  = blockIdx.z;
  const int m0 = blockIdx.y * 128;
  const int n0 = blockIdx.x * 128;
  const int tid  = threadIdx.x;
  const int lane = tid & 31;
  const int wave = tid >> 5;
  const int half = lane >> 4;
  const int l16  = lane & 15;
  const int wm = wave & 1;
  const int wn = wave >> 1;

  v8f acc[4][2];
#pragma unroll
  for (int i = 0; i < 4; ++i)
#pragma unroll
    for (int j = 0; j < 2; ++j) acc[i][j] = (v8f)0.0f;

  const float* Ab = Attn + (size_t)b * 256 * 256;
  const float* Hb = H1 + (size_t)b * 256 * 3136;

  const int jm = tid & 127;
  const int kq = (tid >> 7) * 8;
  const int br = tid >> 4;          // B-tile row 0..15
  const int bc = (tid & 15) * 8;    // B-tile col base

  const float* pa = Ab + (size_t)(m0 + jm) * 256 + kq;

  float ra[8], rb[8];
#pragma unroll
  for (int i = 0; i < 8; ++i) ra[i] = pa[i];
  agg_loadB(Hb, n0, br, bc, 0, rb);
  store_chunk(As[0], kq, jm, ra);
#pragma unroll
  for (int i = 0; i < 8; ++i) Bs[0][br][bc + i] = rb[i];
  __syncthreads();

  int cur = 0;
  const int NC = 256 / KCHUNK;  // 16
  for (int c = 0; c < NC; ++c) {
    const bool more = (c + 1 < NC);
    if (more) {
      const int kn = (c + 1) * KCHUNK;
#pragma unroll
      for (int i = 0; i < 8; ++i) ra[i] = pa[kn + i];
      agg_loadB(Hb, n0, br, bc, kn, rb);
    }
    wmma_block(As[cur], Bs[cur], wm, wn, half, l16, acc);
    if (more) {
      store_chunk(As[cur ^ 1], kq, jm, ra);
#pragma unroll
      for (int i = 0; i < 8; ++i) Bs[cur ^ 1][br][bc + i] = rb[i];
    }
    __syncthreads();
    cur ^= 1;
  }

  float* Ob = Out + (size_t)b * 256 * 3136;
#pragma unroll
  for (int ms = 0; ms < 4; ++ms)
#pragma unroll
    for (int ns = 0; ns < 2; ++ns) {
      const int col = n0 + wn * 32 + ns * 16 + l16;
      if (col < 3136) {
#pragma unroll
        for (int v = 0; v < 8; ++v) {
          const int row = m0 + wm * 64 + ms * 16 + v + 8 * half;
          Ob[(size_t)row * 3136 + col] =
              0.1f * acc[ms][ns][v] + 0.9f * Hb[(size_t)row * 3136 + col];
        }
      }
    }
}

// ---------------------------------------------------------------------------
extern "C" void kernel_launch(void* const* d_in, const int* in_sizes, int n_in,
                              void* d_out, int out_size, void* d_ws, size_t ws_size,
                              hipStream_t stream) {
  const float* x    = (const float*)d_in[0];  // [32,64,56,56]
  const float* nz   = (const float*)d_in[1];  // [32,64,56,56]
  const float* W    = (const float*)d_in[2];  // [256,64,3,3]
  const float* bias = (const float*)d_in[3];  // [256]
  float* out = (float*)d_out;                 // [32,256,56,56]

  const size_t HSZ = (size_t)32 * 256 * 3136;       // h1 / hn element count
  float* h1   = (float*)d_ws;
  float* hn   = h1 + HSZ;
  float* attn = hn + HSZ;                           // [32,256,256]

  dim3 blk(256);
  dim3 convGrid(25, 2, 32);    // N tiles (ceil 3136/128), M tiles, batch
  dim3 scoreGrid(2, 2, 32);    // 256/128 x 256/128 x batch
  dim3 aggGrid(25, 2, 32);

  conv_wmma<<<convGrid, blk, 0, stream>>>(x, W, bias, h1);
  conv_wmma<<<convGrid, blk, 0, stream>>>(nz, W, bias, hn);
  scores_wmma<<<scoreGrid, blk, 0, stream>>>(h1, hn, attn);
  softmax_col<<<dim3(32), blk, 0, stream>>>(attn);
  agg_wmma<<<aggGrid, blk, 0, stream>>>(attn, h1, out);
}